// State_38714835206885
// MI455X (gfx1250) — compile-verified
//
#include <hip/hip_runtime.h>

// ---------------------------------------------------------------------------
// Shapes (fixed by setup_inputs): B=4, M=1088, Hh=32, D=128, F=128, C=512
// Outputs (flat, in order): y_num(4,512,32,128) y_den(4,512,32,1)
//                           K,V,FK(4,1088,32,128) H(4,32,128,128) S(4,32,128)
// ---------------------------------------------------------------------------

typedef float v2f  __attribute__((ext_vector_type(2)));
typedef float v8f  __attribute__((ext_vector_type(8)));
typedef unsigned int u32x4 __attribute__((ext_vector_type(4)));
typedef int   i32x4 __attribute__((ext_vector_type(4)));
typedef int   i32x8 __attribute__((ext_vector_type(8)));

#define BB 4
#define MM 1088
#define HH 32
#define DD 128
#define FF 128
#define CC 512
#define MSTRIDE 4096      // Hh*F == Hh*D elements between consecutive m rows

#if defined(__has_builtin)
#if __has_builtin(__builtin_amdgcn_tensor_load_to_lds) && __has_builtin(__builtin_amdgcn_s_wait_tensorcnt)
#define HAVE_TDM 1
#endif
#endif
#ifndef HAVE_TDM
#define HAVE_TDM 0
#endif

// ---------------------------------------------------------------------------
// Stage `rows` rows of `width=128` f32 (row stride strideElems) from global
// into LDS.  TDM path: build D# per CDNA5 ISA ch.8 and issue
// tensor_load_to_lds (TENSORcnt).  Fallback: plain loads by the calling wave.
// ---------------------------------------------------------------------------
#if HAVE_TDM
__device__ __forceinline__ void tdm_load_2d(unsigned lds_byte_addr,
                                            const float* gaddr,
                                            int rows, int strideElems)
{
    unsigned long long ga = (unsigned long long)gaddr;
    unsigned lo   = (unsigned)__builtin_amdgcn_readfirstlane((int)(unsigned)ga);
    unsigned hi   = (unsigned)__builtin_amdgcn_readfirstlane((int)(unsigned)(ga >> 32));
    unsigned ldsa = (unsigned)__builtin_amdgcn_readfirstlane((int)lds_byte_addr);
    int r         = __builtin_amdgcn_readfirstlane(rows);
    int st        = __builtin_amdgcn_readfirstlane(strideElems);

    u32x4 g0;
    g0[0] = 1u;                                   // count=1 (valid D#)
    g0[1] = ldsa;                                 // lds_addr (bytes)
    g0[2] = lo;                                   // global_addr[31:0]
    g0[3] = (hi & 0x01FFFFFFu) | (2u << 30);      // global_addr[56:32] | type=2

    i32x8 g1;
    g1[0] = 0x00020000;                           // data_size = 4 bytes
    g1[1] = (int)(128u << 16);                    // tensor_dim0 = 128 (bits 79:48)
    g1[2] = (int)((unsigned)r << 16);             // tensor_dim1 = rows (bits 111:80)
    g1[3] = (int)(128u << 16);                    // tile_dim0  = 128 (bits 127:112)
    g1[4] = r;                                    // tile_dim1  = rows (bits 143:128)
    g1[5] = st;                                   // tensor_dim0_stride (bits 207:160)
    g1[6] = 0;
    g1[7] = 0;

    i32x4 z4 = {0, 0, 0, 0};
#if defined(__clang_major__) && __clang_major__ >= 23
    i32x8 z8 = {0, 0, 0, 0, 0, 0, 0, 0};
    __builtin_amdgcn_tensor_load_to_lds(g0, g1, z4, z4, z8, 0);
#else
    __builtin_amdgcn_tensor_load_to_lds(g0, g1, z4, z4, 0);
#endif
}
#endif

__device__ __forceinline__ void stage_rows(float* lds_dst, const float* gsrc,
                                           int rows, int strideElems, int lane)
{
#if HAVE_TDM
    tdm_load_2d((unsigned)(unsigned long long)lds_dst, gsrc, rows, strideElems);
#else
    const int n = rows * 32;                      // float4 count (width 128)
    for (int i = lane; i < n; i += 32) {
        int rr = i >> 5, cc = i & 31;
        ((float4*)lds_dst)[rr * 32 + cc] =
            ((const float4*)(gsrc + (size_t)rr * strideElems))[cc];
    }
#endif
}

__device__ __forceinline__ void dma_wait()
{
#if HAVE_TDM
    __builtin_amdgcn_s_wait_tensorcnt(0);
#endif
}

// ---------------------------------------------------------------------------
// Ring write: dst = src with chunk ring-written at `pos`. float4 vectorized.
// ---------------------------------------------------------------------------
__global__ __launch_bounds__(256) void ring_copy_kernel(
    const float4* __restrict__ src, const float4* __restrict__ chunk,
    float4* __restrict__ dst,
    const int* __restrict__ wp_p, const int* __restrict__ valid_p)
{
    constexpr int rowV = HH * DD / 4;             // 1024 float4 per m-row
    constexpr size_t total = (size_t)BB * MM * rowV;
    size_t i = (size_t)blockIdx.x * blockDim.x + threadIdx.x;
    if (i >= total) return;

    int b     = (int)(i / ((size_t)MM * rowV));
    size_t r  = i - (size_t)b * MM * rowV;
    int m     = (int)(r / rowV);
    int inner = (int)(r % rowV);

    int wp = __builtin_amdgcn_readfirstlane(wp_p[0]);
    int vd = __builtin_amdgcn_readfirstlane(valid_p[0]);
    int ne = vd + CC - MM;                        // num_excess
    int pos = (ne > 0) ? (wp + ne) % MM : (wp + vd) % MM;

    int crel = m - pos; if (crel < 0) crel += MM;
    float4 val;
    if (crel < CC) val = chunk[((size_t)b * CC + crel) * rowV + inner];
    else           val = src[i];
    dst[i] = val;
}

// ---------------------------------------------------------------------------
// S_out[b,h,f] = S_in[b,h,f] + sum_{e<E} FK[b,(wp+e)%M,h,f]
// ---------------------------------------------------------------------------
__global__ __launch_bounds__(256) void s_update_kernel(
    const float* __restrict__ FK, const float* __restrict__ Sin,
    float* __restrict__ Sout,
    const int* __restrict__ wp_p, const int* __restrict__ valid_p)
{
    int i = blockIdx.x * blockDim.x + threadIdx.x;   // 16384 threads
    if (i >= BB * HH * FF) return;
    int f = i & (FF - 1);
    int h = (i >> 7) & (HH - 1);
    int b = i >> 12;

    int wp = __builtin_amdgcn_readfirstlane(wp_p[0]);
    int vd = __builtin_amdgcn_readfirstlane(valid_p[0]);
    int Elen = vd + CC - MM; if (Elen < 0) Elen = 0; if (Elen > CC) Elen = CC;

    const float* base = FK + (size_t)b * MM * MSTRIDE + (size_t)h * FF + f;
    float s = Sin[i];
    for (int e = 0; e < Elen; ++e) {
        int m = wp + e; if (m >= MM) m -= MM;
        s += base[(size_t)m * MSTRIDE];
    }
    Sout[i] = s;
}

// ---------------------------------------------------------------------------
// H_out[b,h] = H_in[b,h] + old_fk^T(FxE) @ old_v(ExD), E = num_excess.
// One block per (b,h): 8 waves, wave w owns f-row w*16 across all 8 d-tiles
// (64 acc VGPRs).  E streamed through double-buffered 16-row LDS panels
// fetched by the Tensor Data Mover (wave 0 issues, TENSORcnt-waited).
// WMMA lane layout per ISA 7.12.2 (A: lane=M, half+VGPR=K; B/D: lane=col).
// ---------------------------------------------------------------------------
__global__ __launch_bounds__(256) void h_update_kernel(
    const float* __restrict__ FK, const float* __restrict__ V,
    const float* __restrict__ Hin, float* __restrict__ Hout,
    const int* __restrict__ wp_p, const int* __restrict__ valid_p)
{
    __shared__ float ldsFK[2][16][FF];            // 16 KB
    __shared__ float ldsV [2][16][DD];            // 16 KB

    const int wp = __builtin_amdgcn_readfirstlane(wp_p[0]);
    const int vd = __builtin_amdgcn_readfirstlane(valid_p[0]);
    int Elen = vd + CC - MM; if (Elen < 0) Elen = 0; if (Elen > CC) Elen = CC;

    const int wave = threadIdx.x >> 5;
    const int lane = threadIdx.x & 31;
    const int halfSel = lane >> 4;
    const int laneM   = lane & 15;
    const int b = blockIdx.x >> 5;                // 128 blocks = (b,h)
    const int h = blockIdx.x & 31;
    const int f0 = wave * 16;

    const float* fkSlab = FK + (size_t)b * MM * MSTRIDE + (size_t)h * FF;
    const float* vSlab  = V  + (size_t)b * MM * MSTRIDE + (size_t)h * DD;

    const size_t hBase = ((size_t)b * HH + h) * (size_t)(FF * DD);
    v8f acc[8];
#pragma unroll
    for (int t = 0; t < 8; ++t)
#pragma unroll
        for (int g = 0; g < 8; ++g)
            acc[t][g] = Hin[hBase + (size_t)(f0 + g + 8 * halfSel) * DD + t * 16 + laneM];

    const int chunks = Elen >> 4;

    if (chunks > 0) {                              // prologue: stage chunk 0
        if (wave == 0) {
            int m0 = wp; if (m0 >= MM) m0 -= MM;
            int r1 = MM - m0; if (r1 > 16) r1 = 16;
            stage_rows(&ldsFK[0][0][0], fkSlab + (size_t)m0 * MSTRIDE, r1, MSTRIDE, lane);
            stage_rows(&ldsV [0][0][0], vSlab  + (size_t)m0 * MSTRIDE, r1, MSTRIDE, lane);
            if (r1 < 16) {                         // ring wrap: second segment
                stage_rows(&ldsFK[0][r1][0], fkSlab, 16 - r1, MSTRIDE, lane);
                stage_rows(&ldsV [0][r1][0], vSlab,  16 - r1, MSTRIDE, lane);
            }
            dma_wait();
        }
        __syncthreads();
    }

    for (int ci = 0; ci < chunks; ++ci) {
        const int cur = ci & 1;
        if (wave == 0 && ci + 1 < chunks) {        // prefetch next chunk
            const int nxt = cur ^ 1;
            int m0 = wp + (ci + 1) * 16; if (m0 >= MM) m0 -= MM;
            int r1 = MM - m0; if (r1 > 16) r1 = 16;
            stage_rows(&ldsFK[nxt][0][0], fkSlab + (size_t)m0 * MSTRIDE, r1, MSTRIDE, lane);
            stage_rows(&ldsV [nxt][0][0], vSlab  + (size_t)m0 * MSTRIDE, r1, MSTRIDE, lane);
            if (r1 < 16) {
                stage_rows(&ldsFK[nxt][r1][0], fkSlab, 16 - r1, MSTRIDE, lane);
                stage_rows(&ldsV [nxt][r1][0], vSlab,  16 - r1, MSTRIDE, lane);
            }
        }
#pragma unroll
        for (int e0 = 0; e0 < 16; e0 += 4) {       // 4 WMMA k-steps per chunk
            v2f a;
            a.x = ldsFK[cur][e0 + 2 * halfSel    ][f0 + laneM];
            a.y = ldsFK[cur][e0 + 2 * halfSel + 1][f0 + laneM];
#pragma unroll
            for (int t = 0; t < 8; ++t) {
                v2f bm;
                bm.x = ldsV[cur][e0 + 2 * halfSel    ][t * 16 + laneM];
                bm.y = ldsV[cur][e0 + 2 * halfSel + 1][t * 16 + laneM];
                acc[t] = __builtin_amdgcn_wmma_f32_16x16x4_f32(
                    false, a, false, bm, (short)0, acc[t], false, false);
            }
        }
        if (wave == 0 && ci + 1 < chunks) dma_wait();
        __syncthreads();
    }

    const int tail = Elen & 15;                    // remainder: direct global
    if (tail) {
        const int eStart = Elen & ~15;
        const float* fkB = fkSlab + f0 + laneM;
        const float* vB  = vSlab + laneM;
        for (int e0 = 0; e0 < tail; e0 += 4) {
            const int ea = e0 + 2 * halfSel;
            v2f a = {0.f, 0.f};
            if (ea < tail)     { int m = wp + eStart + ea;     if (m >= MM) m -= MM;
                                 a.x = fkB[(size_t)m * MSTRIDE]; }
            if (ea + 1 < tail) { int m = wp + eStart + ea + 1; if (m >= MM) m -= MM;
                                 a.y = fkB[(size_t)m * MSTRIDE]; }
#pragma unroll
            for (int t = 0; t < 8; ++t) {
                v2f bm = {0.f, 0.f};
                if (ea < tail)     { int m = wp + eStart + ea;     if (m >= MM) m -= MM;
                                     bm.x = vB[(size_t)m * MSTRIDE + t * 16]; }
                if (ea + 1 < tail) { int m = wp + eStart + ea + 1; if (m >= MM) m -= MM;
                                     bm.y = vB[(size_t)m * MSTRIDE + t * 16]; }
                acc[t] = __builtin_amdgcn_wmma_f32_16x16x4_f32(
                    false, a, false, bm, (short)0, acc[t], false, false);
            }
        }
    }

#pragma unroll
    for (int t = 0; t < 8; ++t)
#pragma unroll
        for (int g = 0; g < 8; ++g)
            Hout[hBase + (size_t)(f0 + g + 8 * halfSel) * DD + t * 16 + laneM] = acc[t][g];
}

// ---------------------------------------------------------------------------
// y_num[b,c,h,d] = fq(CxF) @ H_out(FxD) per (b,h).
// 4 blocks per (b,h); whole 64 KB H[b,h] slab pulled into LDS by the TDM
// once per block; wave per c-tile computes 8 d-tiles; fq A-pairs contiguous.
// ---------------------------------------------------------------------------
__global__ __launch_bounds__(256) void ynum_kernel(
    const float* __restrict__ fq, const float* __restrict__ Hout,
    float* __restrict__ ynum)
{
    __shared__ float ldsH[FF * DD];               // 64 KB

    const int wave = threadIdx.x >> 5;
    const int lane = threadIdx.x & 31;
    const int halfSel = lane >> 4;
    const int laneM   = lane & 15;
    const int bh   = blockIdx.x >> 2;             // 512 blocks
    const int cblk = blockIdx.x & 3;
    const int b = bh >> 5, h = bh & 31;
    const int c0 = cblk * 128 + wave * 16;

    const size_t hBase = ((size_t)b * HH + h) * (size_t)(FF * DD);
    if (wave == 0) {
        stage_rows(ldsH, Hout + hBase, 128, 128, lane);
        dma_wait();
    }
    __syncthreads();

    const float* fqB = fq + ((size_t)b * CC + c0 + laneM) * MSTRIDE
                          + (size_t)h * FF + 2 * halfSel;

    v8f acc[8] = {};
#pragma unroll 2
    for (int f0v = 0; f0v < FF; f0v += 4) {
        if (f0v + 32 < FF)
            __builtin_prefetch(fqB + f0v + 32, 0, 1);
        v2f a;
        a.x = fqB[f0v];
        a.y = fqB[f0v + 1];
        const float* hrow = &ldsH[(f0v + 2 * halfSel) * DD + laneM];
#pragma unroll
        for (int t = 0; t < 8; ++t) {
            v2f bm;
            bm.x = hrow[t * 16];
            bm.y = hrow[DD + t * 16];
            acc[t] = __builtin_amdgcn_wmma_f32_16x16x4_f32(
                false, a, false, bm, (short)0, acc[t], false, false);
        }
    }

    // D tile row c = c0 + g + 8*halfSel, col d = t*16 + laneM; y:(B,C,H,D)
    size_t yBase = (((size_t)b * CC + c0 + 8 * halfSel) * HH + h) * DD + laneM;
#pragma unroll
    for (int t = 0; t < 8; ++t)
#pragma unroll
        for (int g = 0; g < 8; ++g)
            ynum[yBase + (size_t)g * (HH * DD) + t * 16] = acc[t][g];
}

// ---------------------------------------------------------------------------
// y_den[b,c,h] = dot(fq[b,c,h,:], S_out[b,h,:]) — one wave32 per output.
// ---------------------------------------------------------------------------
__global__ __launch_bounds__(256) void yden_kernel(
    const float* __restrict__ fq, const float* __restrict__ Sout,
    float* __restrict__ yden)
{
    int wave = (blockIdx.x * blockDim.x + threadIdx.x) >> 5;   // (b,c,h)
    int lane = threadIdx.x & 31;
    int h = wave & 31;
    int c = (wave >> 5) & (CC - 1);
    int b = wave >> 14;

    const float* fqp = fq + (((size_t)b * CC + c) * HH + h) * FF;
    const float* sp  = Sout + ((size_t)b * HH + h) * FF;
    float acc = 0.f;
#pragma unroll
    for (int k = 0; k < FF; k += 32)
        acc += fqp[k + lane] * sp[k + lane];
    for (int off = 16; off > 0; off >>= 1)
        acc += __shfl_down(acc, off, 32);
    if (lane == 0) yden[wave] = acc;
}

// ---------------------------------------------------------------------------
extern "C" void kernel_launch(void* const* d_in, const int* in_sizes, int n_in,
                              void* d_out, int out_size, void* d_ws, size_t ws_size,
                              hipStream_t stream)
{
    const float* K    = (const float*)d_in[0];
    const float* V    = (const float*)d_in[1];
    const float* FK   = (const float*)d_in[2];
    const float* H    = (const float*)d_in[3];
    const float* S    = (const float*)d_in[4];
    const float* k_c  = (const float*)d_in[5];
    const float* v_c  = (const float*)d_in[6];
    const float* fk_c = (const float*)d_in[7];
    const float* fq   = (const float*)d_in[8];
    const int*   wp   = (const int*)d_in[9];
    const int*   vld  = (const int*)d_in[10];

    float* out   = (float*)d_out;
    float* ynum  = out;                                    //  8,388,608
    float* yden  = out + (size_t)8388608;                  //     65,536
    float* Kout  = out + (size_t)8454144;                  // 17,825,792
    float* Vout  = out + (size_t)26279936;                 // 17,825,792
    float* FKout = out + (size_t)44105728;                 // 17,825,792
    float* Hout  = out + (size_t)61931520;                 //  2,097,152
    float* Sout  = out + (size_t)64028672;                 //     16,384

    {   // Ring writes of K/V/FK (float4: 4*1088*1024 = 4,456,448 vectors)
        dim3 blk(256), grd(17408);
        ring_copy_kernel<<<grd, blk, 0, stream>>>((const float4*)K,  (const float4*)k_c,  (float4*)Kout,  wp, vld);
        ring_copy_kernel<<<grd, blk, 0, stream>>>((const float4*)V,  (const float4*)v_c,  (float4*)Vout,  wp, vld);
        ring_copy_kernel<<<grd, blk, 0, stream>>>((const float4*)FK, (const float4*)fk_c, (float4*)FKout, wp, vld);
    }
    // S update (16384 threads)
    s_update_kernel<<<64, 256, 0, stream>>>(FK, S, Sout, wp, vld);
    // H update: one block per (b,h) = 128 blocks x 256 threads
    h_update_kernel<<<128, 256, 0, stream>>>(FK, V, H, Hout, wp, vld);
    // y_den: 65536 waves -> 8192 blocks of 256
    yden_kernel<<<8192, 256, 0, stream>>>(fq, Sout, yden);
    // y_num: 4 blocks per (b,h) = 512 blocks x 256 threads
    ynum_kernel<<<512, 256, 0, stream>>>(fq, Hout, ynum);
}